// S4Block_3994319585995
// MI455X (gfx1250) — compile-verified
//
#include <hip/hip_runtime.h>
#include <hip/hip_bf16.h>
#include <math.h>

// ---------------------------------------------------------------------------
// S4 block for MI455X (gfx1250, wave32).
//   D_MODEL=256, N_STATE=64, L=4096, BATCH=8, all float32.
// Pipeline:
//   1) transpose W -> Wt (for coalesced WMMA B-fragments)
//   2) transpose u[B,L,D] -> ut[B,D,L] (contiguous FFT rows)
//   3) kd_kernel:  Cauchy sum -> ifft(4096) -> zero-pad -> fft(8192) => Kd[d][8192] complex
//   4) conv_kernel: per (b,d) row: fft(8192) in LDS, *Kd, ifft, +D_skip*u, exact GELU => zt
//   5) linear_ln_kernel: z @ W^T + bias (V_WMMA_F32_16X16X4_F32), +u residual, LayerNorm
// Workspace usage: Wt(256KB) + ut(32MB) + Kd(16MB) + zt(32MB) ~= 80.3 MB.
// ---------------------------------------------------------------------------

#define LSEQ   4096
#define L2SEQ  8192
#define DM     256
#define NST    64
#define BSZ    8
#define PI_F   3.14159265358979323846f

typedef __attribute__((ext_vector_type(2))) float v2f;
typedef __attribute__((ext_vector_type(8))) float v8f;

__device__ __forceinline__ float2 mkc(float a, float b) { float2 r; r.x = a; r.y = b; return r; }
__device__ __forceinline__ float2 cmulf(float2 a, float2 b) {
  return mkc(a.x * b.x - a.y * b.y, a.x * b.y + a.y * b.x);
}
__device__ __forceinline__ float2 cfmaf(float2 a, float2 b, float2 c) {
  return mkc(fmaf(a.x, b.x, fmaf(-a.y, b.y, c.x)), fmaf(a.x, b.y, fmaf(a.y, b.x, c.y)));
}
__device__ __forceinline__ float2 crecipf(float2 z) {
  float inv = 1.0f / fmaf(z.x, z.x, z.y * z.y);
  return mkc(z.x * inv, -z.y * inv);
}

// In-place radix-2 DIT FFT on LDS data. n = 4096 or 8192, 256 threads.
// dirSign = -1 forward, +1 inverse (no 1/n scaling applied here).
__device__ __forceinline__ void fft_lds(float2* s, int n, int logn, int tid, float dirSign) {
  __syncthreads();
  for (int i = tid; i < n; i += 256) {
    unsigned r = __brev((unsigned)i) >> (32 - logn);
    if (r > (unsigned)i) { float2 t = s[i]; s[i] = s[r]; s[r] = t; }
  }
  __syncthreads();
  int sh = 0;
  for (int len = 2; len <= n; len <<= 1, ++sh) {
    int half = 1 << sh;
    float base = dirSign * 2.0f * PI_F / (float)len;
    int nbf = n >> 1;
    for (int bf = tid; bf < nbf; bf += 256) {
      int j  = bf & (half - 1);
      int i0 = ((bf >> sh) << (sh + 1)) | j;
      int i1 = i0 + half;
      float sn, cs;
      __sincosf(base * (float)j, &sn, &cs);
      float2 a = s[i0], b = s[i1];
      float tr = cs * b.x - sn * b.y;
      float ti = cs * b.y + sn * b.x;
      s[i0] = mkc(a.x + tr, a.y + ti);
      s[i1] = mkc(a.x - tr, a.y - ti);
    }
    __syncthreads();
  }
}

// ---------------------------------------------------------------------------
// Batched tile transpose: in[b][r][c] -> out[b][c][r].  block (32,8).
// ---------------------------------------------------------------------------
__global__ __launch_bounds__(256) void transpose_kernel(const float* __restrict__ in,
                                                        float* __restrict__ out,
                                                        int R, int C) {
  __shared__ float tile[32][33];
  int b = blockIdx.z;
  const float* inb = in + (size_t)b * R * C;
  float* outb = out + (size_t)b * R * C;
  int c0 = blockIdx.x * 32, r0 = blockIdx.y * 32;
  int tx = threadIdx.x, ty = threadIdx.y;
#pragma unroll
  for (int i = 0; i < 32; i += 8)
    tile[ty + i][tx] = inb[(size_t)(r0 + ty + i) * C + (c0 + tx)];
  __syncthreads();
#pragma unroll
  for (int i = 0; i < 32; i += 8)
    outb[(size_t)(c0 + ty + i) * R + (r0 + tx)] = tile[tx][ty + i];
}

// ---------------------------------------------------------------------------
// Kernel K-hat builder. One workgroup per d (256 WGs, 256 threads, 64KB LDS).
//   at_roots[l] = c_l * (k00 - k01*k10/(1+k11)) over N=64 poles
//   K = real(ifft_4096(at_roots));  Kd = fft_8192(zero-pad(K))
// Coefficient tables live in the (initially unused) upper half of the LDS buf.
// ---------------------------------------------------------------------------
__global__ __launch_bounds__(256) void kd_kernel(const float* __restrict__ lam_ri,
                                                 const float* __restrict__ p_ri,
                                                 const float* __restrict__ q_ri,
                                                 const float* __restrict__ B_ri,
                                                 const float* __restrict__ Ct_ri,
                                                 const float* __restrict__ log_step,
                                                 float* __restrict__ Kd) {
  __shared__ float2 s[L2SEQ];           // 64 KB
  float2* aA = s + 4096;                // conj(Ct)*B
  float2* bA = s + 4160;                // conj(Ct)*p
  float2* cA = s + 4224;                // conj(q)*B
  float2* wA = s + 4288;                // conj(q)*p
  float2* lA = s + 4352;                // lambda
  int d = blockIdx.x, t = threadIdx.x;

  if (t < NST) {
    int n = t;
    float2 lam = mkc(lam_ri[2 * n], lam_ri[2 * n + 1]);
    float2 p   = mkc(p_ri[2 * n],   p_ri[2 * n + 1]);
    float2 q   = mkc(q_ri[2 * n],   q_ri[2 * n + 1]);
    float2 Bd  = mkc(B_ri[(d * NST + n) * 2],  B_ri[(d * NST + n) * 2 + 1]);
    float2 Ct  = mkc(Ct_ri[(d * NST + n) * 2], Ct_ri[(d * NST + n) * 2 + 1]);
    float2 Ctc = mkc(Ct.x, -Ct.y);
    float2 qc  = mkc(q.x,  -q.y);
    aA[n] = cmulf(Ctc, Bd);
    bA[n] = cmulf(Ctc, p);
    cA[n] = cmulf(qc, Bd);
    wA[n] = cmulf(qc, p);
    lA[n] = lam;
  }
  __syncthreads();

  float gscale = 2.0f / expf(log_step[d]);
  for (int l = t; l < LSEQ; l += 256) {
    float theta = -2.0f * PI_F * (float)l / (float)LSEQ;
    float sn, cs;
    __sincosf(theta, &sn, &cs);
    float2 om    = mkc(cs, sn);
    float2 onep  = mkc(1.0f + om.x,  om.y);
    float2 onem  = mkc(1.0f - om.x, -om.y);
    float2 inv1p = crecipf(onep);
    float2 cval  = mkc(2.0f * inv1p.x, 2.0f * inv1p.y);
    float2 gm    = cmulf(onem, inv1p);
    float2 g     = mkc(gscale * gm.x, gscale * gm.y);
    float2 k00 = mkc(0, 0), k01 = mkc(0, 0), k10 = mkc(0, 0), k11 = mkc(0, 0);
#pragma unroll 8
    for (int n = 0; n < NST; ++n) {
      float2 lam = lA[n];
      float2 r = crecipf(mkc(g.x - lam.x, g.y - lam.y));
      k00 = cfmaf(aA[n], r, k00);
      k01 = cfmaf(bA[n], r, k01);
      k10 = cfmaf(cA[n], r, k10);
      k11 = cfmaf(wA[n], r, k11);
    }
    float2 onek = mkc(1.0f + k11.x, k11.y);
    float2 tmp  = cmulf(cmulf(k01, crecipf(onek)), k10);
    s[l] = cmulf(cval, mkc(k00.x - tmp.x, k00.y - tmp.y));
  }

  fft_lds(s, LSEQ, 12, t, +1.0f);       // inverse FFT (unscaled)

  // expand: K (real, scaled 1/L) into zero-padded length-2L complex buffer
  for (int l = t; l < LSEQ; l += 256)
    s[l] = mkc(s[l].x * (1.0f / (float)LSEQ), 0.0f);
  for (int l = LSEQ + t; l < L2SEQ; l += 256)
    s[l] = mkc(0.0f, 0.0f);

  fft_lds(s, L2SEQ, 13, t, -1.0f);      // forward FFT of padded K

  float* kdrow = Kd + (size_t)d * L2SEQ * 2;
  for (int f = t; f < L2SEQ; f += 256) {
    kdrow[2 * f]     = s[f].x;
    kdrow[2 * f + 1] = s[f].y;
  }
}

// ---------------------------------------------------------------------------
// Fused conv row: fft(u-pad) * Kd -> ifft -> +D_skip*u -> exact GELU => zt.
// One workgroup per (b,d): 2048 WGs, 256 threads, 64KB LDS.
// ---------------------------------------------------------------------------
__global__ __launch_bounds__(256) void conv_kernel(const float* __restrict__ ut,
                                                   const float* __restrict__ Kd,
                                                   const float* __restrict__ D_skip,
                                                   float* __restrict__ zt) {
  __shared__ float2 s[L2SEQ];           // 64 KB
  int blk = blockIdx.x;
  int b = blk >> 8, d = blk & 255;
  int t = threadIdx.x;
  const float* urow = ut + ((size_t)b * DM + d) * LSEQ;

  for (int l = t; l < LSEQ; l += 256)       s[l] = mkc(urow[l], 0.0f);
  for (int l = LSEQ + t; l < L2SEQ; l += 256) s[l] = mkc(0.0f, 0.0f);

  fft_lds(s, L2SEQ, 13, t, -1.0f);

  const float* kdrow = Kd + (size_t)d * L2SEQ * 2;
  const float scale = 1.0f / (float)L2SEQ;  // ifft normalization folded in
  for (int f = t; f < L2SEQ; f += 256) {
    float2 k = mkc(kdrow[2 * f], kdrow[2 * f + 1]);
    float2 pr = cmulf(s[f], k);
    s[f] = mkc(pr.x * scale, pr.y * scale);
  }

  fft_lds(s, L2SEQ, 13, t, +1.0f);

  float ds = D_skip[d];
  float* zrow = zt + ((size_t)b * DM + d) * LSEQ;
  for (int l = t; l < LSEQ; l += 256) {
    float z = fmaf(ds, urow[l], s[l].x);
    zrow[l] = 0.5f * z * (1.0f + erff(z * 0.70710678118654752f));  // exact GELU
  }
}

// ---------------------------------------------------------------------------
// z @ W^T + bias, residual, LayerNorm with V_WMMA_F32_16X16X4_F32.
// One workgroup per (b, 16-row l-tile): 2048 WGs, 256 threads (8 waves).
// Wave w computes N-tiles j0 = 32w and 32w+16; K=256 in 64 steps of 4.
// ISA 7.12.2 f32 layouts: A 16x4 -> lanes0-15:{K0,K1}, lanes16-31:{K2,K3};
// B 4x16 mirrored; C/D: vgpr r -> row r (lanes0-15) / r+8 (lanes16-31).
// ---------------------------------------------------------------------------
#define ZPAD 257
__global__ __launch_bounds__(256) void linear_ln_kernel(const float* __restrict__ zt,
                                                        const float* __restrict__ Wt,
                                                        const float* __restrict__ bias,
                                                        const float* __restrict__ u,
                                                        const float* __restrict__ gamma,
                                                        const float* __restrict__ beta,
                                                        float* __restrict__ out) {
  __shared__ float zs[16 * ZPAD];   // A tile, padded against bank conflicts
  __shared__ float ys[16 * ZPAD];   // y tile for LN
  int blk  = blockIdx.x;
  int b    = blk >> 8;
  int l0   = (blk & 255) * 16;
  int t    = threadIdx.x;

  // stage z tile: zs[m][k] = zt[b][k][l0+m]  (16 consecutive floats per k-row)
  for (int idx = t; idx < 16 * DM; idx += 256) {
    int m = idx & 15, k = idx >> 4;
    zs[m * ZPAD + k] = zt[((size_t)b * DM + k) * LSEQ + l0 + m];
  }
  __syncthreads();

  int wave  = t >> 5, lane = t & 31;
  int lmod  = lane & 15;
  int khalf = (lane < 16) ? 0 : 2;   // lanes 16-31 carry K+2 per ISA layout
  int mrow  = lmod;                  // A-matrix row for this lane

#pragma unroll
  for (int rep = 0; rep < 2; ++rep) {
    int j0 = (wave * 2 + rep) * 16;
    v8f acc = {};
    for (int k0 = 0; k0 < DM; k0 += 4) {
      v2f A, Bf;
      A.x  = zs[mrow * ZPAD + k0 + khalf];
      A.y  = zs[mrow * ZPAD + k0 + khalf + 1];
      Bf.x = Wt[(size_t)(k0 + khalf) * DM + j0 + lmod];      // W^T row: coalesced
      Bf.y = Wt[(size_t)(k0 + khalf + 1) * DM + j0 + lmod];
      acc = __builtin_amdgcn_wmma_f32_16x16x4_f32(
          /*neg_a=*/false, A, /*neg_b=*/false, Bf,
          /*c_mod=*/(short)0, acc, /*reuse_a=*/false, /*reuse_b=*/false);
    }
    int mbase = (lane < 16) ? 0 : 8;
#pragma unroll
    for (int r = 0; r < 8; ++r)
      ys[(mbase + r) * ZPAD + j0 + lmod] = acc[r];
  }
  __syncthreads();

  // bias + residual + LayerNorm: 16 lanes per row, width-16 shuffle reduce
  int row = t >> 4;
  int sub = t & 15;
  const float* urow = u + ((size_t)b * LSEQ + l0 + row) * DM;
  float sum = 0.0f, sq = 0.0f;
#pragma unroll
  for (int i = 0; i < 16; ++i) {
    int j = sub + i * 16;
    float x = ys[row * ZPAD + j] + bias[j] + urow[j];
    ys[row * ZPAD + j] = x;
    sum += x;
    sq  = fmaf(x, x, sq);
  }
#pragma unroll
  for (int off = 8; off >= 1; off >>= 1) {
    sum += __shfl_xor(sum, off, 16);
    sq  += __shfl_xor(sq,  off, 16);
  }
  float mean = sum * (1.0f / DM);
  float var  = fmaf(-mean, mean, sq * (1.0f / DM));
  float rstd = rsqrtf(var + 1e-5f);
  float* orow = out + ((size_t)b * LSEQ + l0 + row) * DM;
#pragma unroll
  for (int i = 0; i < 16; ++i) {
    int j = sub + i * 16;
    float x = ys[row * ZPAD + j];
    orow[j] = fmaf(gamma[j], (x - mean) * rstd, beta[j]);
  }
}

// ---------------------------------------------------------------------------
extern "C" void kernel_launch(void* const* d_in, const int* in_sizes, int n_in,
                              void* d_out, int out_size, void* d_ws, size_t ws_size,
                              hipStream_t stream) {
  (void)in_sizes; (void)n_in; (void)out_size; (void)ws_size;
  const float* u        = (const float*)d_in[0];
  const float* B_ri     = (const float*)d_in[1];
  const float* Ct_ri    = (const float*)d_in[2];
  const float* lam_ri   = (const float*)d_in[3];
  const float* p_ri     = (const float*)d_in[4];
  const float* q_ri     = (const float*)d_in[5];
  const float* log_step = (const float*)d_in[6];
  const float* D_skip   = (const float*)d_in[7];
  const float* W        = (const float*)d_in[8];
  const float* bias     = (const float*)d_in[9];
  const float* gamma    = (const float*)d_in[10];
  const float* beta     = (const float*)d_in[11];
  float* out = (float*)d_out;

  float* ws = (float*)d_ws;
  float* Wt = ws;                                   // 256*256
  float* ut = Wt + (size_t)DM * DM;                 // B*D*L
  float* Kd = ut + (size_t)BSZ * DM * LSEQ;         // D*8192*2
  float* zt = Kd + (size_t)DM * L2SEQ * 2;          // B*D*L

  dim3 tb(32, 8);
  transpose_kernel<<<dim3(DM / 32, DM / 32, 1), tb, 0, stream>>>(W, Wt, DM, DM);
  transpose_kernel<<<dim3(DM / 32, LSEQ / 32, BSZ), tb, 0, stream>>>(u, ut, LSEQ, DM);
  kd_kernel<<<DM, 256, 0, stream>>>(lam_ri, p_ri, q_ri, B_ri, Ct_ri, log_step, Kd);
  conv_kernel<<<BSZ * DM, 256, 0, stream>>>(ut, Kd, D_skip, zt);
  linear_ln_kernel<<<BSZ * LSEQ / 16, 256, 0, stream>>>(zt, Wt, bias, u, gamma, beta, out);
}